// GAT_86199993631335
// MI455X (gfx1250) — compile-verified
//
#include <hip/hip_runtime.h>
#include <hip/hip_bf16.h>
#include <math.h>

#define N_NODES 100000
#define N_EDGES 1600000
#define IN_CH   512
#define HID     128   // HEADS * OUT_CH
#define HEADS   8
#define OUT_CH  16
#define NEG_SLOPE 0.2f
#define EPS_F    1e-16f

typedef float v2f __attribute__((ext_vector_type(2)));
typedef float v8f __attribute__((ext_vector_type(8)));

// ---- ordered-uint mapping for float atomic max (monotonic, exact) ----
__device__ __forceinline__ unsigned f2key(float f) {
  unsigned b = __float_as_uint(f);
  return (b & 0x80000000u) ? ~b : (b | 0x80000000u);
}
__device__ __forceinline__ float key2f(unsigned k) {
  return __uint_as_float((k & 0x80000000u) ? (k & 0x7fffffffu) : ~k);
}
#define KEY_NEG_INF 0x007FFFFFu   // f2key(-inf)

__device__ __forceinline__ void atomAddF(float* p, float v) {
  unsafeAtomicAdd(p, v);   // native global_atomic_add_f32 on gfx1250
}

// =====================================================================
// GEMM: C[M x 128] = A[M x K] @ B[K x 128], fp32, WMMA 16x16x4_f32
// Block = 8 waves; wave w owns rows [blk*128 + w*16, +16), all 128 cols.
// B tile (64 x 128) staged into LDS with the gfx1250 async global->LDS
// path (no VGPR round-trip, tracked by ASYNCcnt), padded stride 136
// dwords so the two lane-halves hit disjoint bank sets.
// =====================================================================
#define TILE_K 64
#define BPAD   136

__global__ __launch_bounds__(256) void gat_gemm(
    const float* __restrict__ A, const float* __restrict__ B,
    float* __restrict__ C, int M, int K)
{
  __shared__ float Bs[TILE_K * BPAD];
  const int lane    = threadIdx.x & 31;
  const int wave    = threadIdx.x >> 5;
  const int l16     = lane & 15;
  const int halfsel = lane >> 4;          // 0: lanes 0-15, 1: lanes 16-31
  const int row0    = (blockIdx.x * 8 + wave) * 16;

  int row = row0 + l16;
  if (row >= M) row = M - 1;              // clamped dup rows never stored

  v8f acc[8];
  #pragma unroll
  for (int t = 0; t < 8; t++)
    #pragma unroll
    for (int j = 0; j < 8; j++) acc[t][j] = 0.0f;

  for (int kc = 0; kc < K; kc += TILE_K) {
    // async stage of B[kc..kc+63][0..127] -> LDS, 16B per lane per issue
    #pragma unroll
    for (int i = threadIdx.x; i < TILE_K * 32; i += 256) {
      int r = i >> 5, c4 = (i & 31) * 4;
      const float* gsrc = B + (size_t)(kc + r) * HID + c4;
      unsigned lds_off = (unsigned)(size_t)(&Bs[r * BPAD + c4]);
      asm volatile("global_load_async_to_lds_b128 %0, %1, off"
                   :: "v"(lds_off), "v"(gsrc) : "memory");
    }
    asm volatile("s_wait_asynccnt 0x0" ::: "memory");
    __syncthreads();

    #pragma unroll 4
    for (int k2 = 0; k2 < TILE_K; k2 += 4) {
      const int kk = k2 + 2 * halfsel;    // A/B fragment K slot for this half
      v2f a;
      const float* ap = A + (size_t)row * K + kc + kk;
      a[0] = ap[0];
      a[1] = ap[1];
      #pragma unroll
      for (int t = 0; t < 8; t++) {
        v2f b;
        b[0] = Bs[kk * BPAD + t * 16 + l16];
        b[1] = Bs[(kk + 1) * BPAD + t * 16 + l16];
        acc[t] = __builtin_amdgcn_wmma_f32_16x16x4_f32(
            false, a, false, b, (short)0, acc[t], false, false);
      }
    }
    __syncthreads();
  }

  // C/D layout: VGPR r -> row r (lanes 0-15) / row r+8 (lanes 16-31)
  #pragma unroll
  for (int t = 0; t < 8; t++)
    #pragma unroll
    for (int r = 0; r < 8; r++) {
      int orow = row0 + r + 8 * halfsel;
      if (orow < M) C[(size_t)orow * HID + t * 16 + l16] = acc[t][r];
    }
}

// =====================================================================
// edge index int64 -> int32
// =====================================================================
__global__ void conv_edges(const long long* __restrict__ ei,
                           int* __restrict__ src, int* __restrict__ dst)
{
  int i = blockIdx.x * blockDim.x + threadIdx.x;
  if (i >= N_EDGES) return;
  src[i] = (int)ei[i];
  dst[i] = (int)ei[(size_t)N_EDGES + i];
}

__global__ void zero_f(float* __restrict__ p, int n)
{
  int i = blockIdx.x * blockDim.x + threadIdx.x;
  if (i < n) p[i] = 0.0f;
}

// =====================================================================
// per (node, head): attention logits + init amax/denom
// =====================================================================
__global__ void gat_prep(const float* __restrict__ h,
                         const float* __restrict__ att_src,
                         const float* __restrict__ att_dst,
                         float* __restrict__ a_src, float* __restrict__ a_dst,
                         unsigned* __restrict__ amax, float* __restrict__ denom)
{
  int tid = blockIdx.x * blockDim.x + threadIdx.x;
  if (tid >= N_NODES * HEADS) return;
  int node = tid >> 3, hd = tid & 7;
  const float4* hp = (const float4*)(h + (size_t)node * HID + hd * OUT_CH);
  const float4* as = (const float4*)(att_src + hd * OUT_CH);
  const float4* ad = (const float4*)(att_dst + hd * OUT_CH);
  float ssum = 0.f, dsum = 0.f;
  #pragma unroll
  for (int j = 0; j < 4; j++) {
    float4 hv = hp[j], sv = as[j], dv = ad[j];
    ssum += hv.x * sv.x + hv.y * sv.y + hv.z * sv.z + hv.w * sv.w;
    dsum += hv.x * dv.x + hv.y * dv.y + hv.z * dv.z + hv.w * dv.w;
  }
  a_src[tid] = ssum;
  a_dst[tid] = dsum;
  amax[tid]  = KEY_NEG_INF;
  denom[tid] = 0.0f;
}

// =====================================================================
// edge pass 1: segment max of leaky_relu(a_src[s] + a_dst[d]) over dst
// =====================================================================
__global__ void gat_edge_max(const int* __restrict__ src, const int* __restrict__ dst,
                             const float* __restrict__ a_src,
                             const float* __restrict__ a_dst,
                             unsigned* __restrict__ amax)
{
  int e = blockIdx.x * blockDim.x + threadIdx.x;
  if (e >= N_EDGES) return;
  int s = src[e], d = dst[e];
  const float4* sp = (const float4*)(a_src + (size_t)s * HEADS);
  const float4* dp = (const float4*)(a_dst + (size_t)d * HEADS);
  float4 s0 = sp[0], s1 = sp[1], d0 = dp[0], d1 = dp[1];
  float al[8] = { s0.x + d0.x, s0.y + d0.y, s0.z + d0.z, s0.w + d0.w,
                  s1.x + d1.x, s1.y + d1.y, s1.z + d1.z, s1.w + d1.w };
  #pragma unroll
  for (int hd = 0; hd < 8; hd++) {
    float a = al[hd];
    a = a > 0.f ? a : a * NEG_SLOPE;
    atomicMax(&amax[(size_t)d * HEADS + hd], f2key(a));
  }
}

// =====================================================================
// edge pass 2: denom[d] += exp(alpha - amax[d])
// =====================================================================
__global__ void gat_edge_sum(const int* __restrict__ src, const int* __restrict__ dst,
                             const float* __restrict__ a_src,
                             const float* __restrict__ a_dst,
                             const unsigned* __restrict__ amax,
                             float* __restrict__ denom)
{
  int e = blockIdx.x * blockDim.x + threadIdx.x;
  if (e >= N_EDGES) return;
  int s = src[e], d = dst[e];
  const float4* sp = (const float4*)(a_src + (size_t)s * HEADS);
  const float4* dp = (const float4*)(a_dst + (size_t)d * HEADS);
  float4 s0 = sp[0], s1 = sp[1], d0 = dp[0], d1 = dp[1];
  float al[8] = { s0.x + d0.x, s0.y + d0.y, s0.z + d0.z, s0.w + d0.w,
                  s1.x + d1.x, s1.y + d1.y, s1.z + d1.z, s1.w + d1.w };
  #pragma unroll
  for (int hd = 0; hd < 8; hd++) {
    float a = al[hd];
    a = a > 0.f ? a : a * NEG_SLOPE;
    float ex = expf(a - key2f(amax[(size_t)d * HEADS + hd]));
    atomAddF(&denom[(size_t)d * HEADS + hd], ex);
  }
}

// =====================================================================
// edge pass 3 (layer 1, concat): out[d, h*16+c] += coef * h1[s, h*16+c]
// one wave32 per edge, lane owns 4 channels
// =====================================================================
__global__ __launch_bounds__(256) void gat_edge_agg_concat(
    const int* __restrict__ src, const int* __restrict__ dst,
    const float* __restrict__ a_src, const float* __restrict__ a_dst,
    const unsigned* __restrict__ amax, const float* __restrict__ denom,
    const float* __restrict__ h, float* __restrict__ out)
{
  int wave = threadIdx.x >> 5, lane = threadIdx.x & 31;
  int e = blockIdx.x * 8 + wave;
  if (e >= N_EDGES) return;
  int s = src[e], d = dst[e];
  int hd = lane >> 2;  // 4 lanes per head
  float a = a_src[(size_t)s * HEADS + hd] + a_dst[(size_t)d * HEADS + hd];
  a = a > 0.f ? a : a * NEG_SLOPE;
  float m  = key2f(amax[(size_t)d * HEADS + hd]);
  float ex = expf(a - m);
  float coef = ex / (denom[(size_t)d * HEADS + hd] + EPS_F);
  float4 hv = *(const float4*)(h + (size_t)s * HID + lane * 4);
  float* op = out + (size_t)d * HID + lane * 4;
  atomAddF(op + 0, hv.x * coef);
  atomAddF(op + 1, hv.y * coef);
  atomAddF(op + 2, hv.z * coef);
  atomAddF(op + 3, hv.w * coef);
}

// =====================================================================
// edge pass 3 (layer 2, mean over heads): out2[d,c] += (1/8)*sum_h coef_h*h2[s,h,c]
// xor-shuffle collapses the 8 heads -> only 16 atomics per edge
// =====================================================================
__global__ __launch_bounds__(256) void gat_edge_agg_mean(
    const int* __restrict__ src, const int* __restrict__ dst,
    const float* __restrict__ a_src, const float* __restrict__ a_dst,
    const unsigned* __restrict__ amax, const float* __restrict__ denom,
    const float* __restrict__ h, float* __restrict__ out2)
{
  int wave = threadIdx.x >> 5, lane = threadIdx.x & 31;
  int e = blockIdx.x * 8 + wave;
  if (e >= N_EDGES) return;
  int s = src[e], d = dst[e];
  int hd = lane >> 2;
  float a = a_src[(size_t)s * HEADS + hd] + a_dst[(size_t)d * HEADS + hd];
  a = a > 0.f ? a : a * NEG_SLOPE;
  float m  = key2f(amax[(size_t)d * HEADS + hd]);
  float ex = expf(a - m);
  float coef = 0.125f * ex / (denom[(size_t)d * HEADS + hd] + EPS_F);
  float4 hv = *(const float4*)(h + (size_t)s * HID + lane * 4);
  float v[4] = { hv.x * coef, hv.y * coef, hv.z * coef, hv.w * coef };
  #pragma unroll
  for (int j = 0; j < 4; j++) {
    v[j] += __shfl_xor(v[j], 4);
    v[j] += __shfl_xor(v[j], 8);
    v[j] += __shfl_xor(v[j], 16);
  }
  if (lane < 4) {
    float* op = out2 + (size_t)d * OUT_CH + lane * 4;
    atomAddF(op + 0, v[0]);
    atomAddF(op + 1, v[1]);
    atomAddF(op + 2, v[2]);
    atomAddF(op + 3, v[3]);
  }
}

// bias + ELU in place (layer-1 output -> layer-2 features)
__global__ void gat_bias_elu(float* __restrict__ io, const float* __restrict__ bias, int n)
{
  int i = blockIdx.x * blockDim.x + threadIdx.x;
  if (i >= n) return;
  float v = io[i] + bias[i & (HID - 1)];
  io[i] = v > 0.f ? v : (expf(v) - 1.f);
}

// bias + log_softmax over 16 channels; one thread per node
__global__ void gat_logsoftmax(const float* __restrict__ out2,
                               const float* __restrict__ bias,
                               float* __restrict__ y)
{
  int node = blockIdx.x * blockDim.x + threadIdx.x;
  if (node >= N_NODES) return;
  float v[16];
  const float4* p = (const float4*)(out2 + (size_t)node * OUT_CH);
  #pragma unroll
  for (int j = 0; j < 4; j++) {
    float4 t = p[j];
    v[4 * j + 0] = t.x + bias[4 * j + 0];
    v[4 * j + 1] = t.y + bias[4 * j + 1];
    v[4 * j + 2] = t.z + bias[4 * j + 2];
    v[4 * j + 3] = t.w + bias[4 * j + 3];
  }
  float m = v[0];
  #pragma unroll
  for (int c = 1; c < 16; c++) m = fmaxf(m, v[c]);
  float ssum = 0.f;
  #pragma unroll
  for (int c = 0; c < 16; c++) ssum += expf(v[c] - m);
  float lse = m + logf(ssum);
  #pragma unroll
  for (int c = 0; c < 16; c++) y[(size_t)node * OUT_CH + c] = v[c] - lse;
}

// =====================================================================
extern "C" void kernel_launch(void* const* d_in, const int* in_sizes, int n_in,
                              void* d_out, int out_size, void* d_ws, size_t ws_size,
                              hipStream_t stream)
{
  (void)in_sizes; (void)n_in; (void)out_size; (void)ws_size;
  const float*      x        = (const float*)d_in[0];
  const long long*  ei       = (const long long*)d_in[1];
  const float*      W1       = (const float*)d_in[2];
  const float*      att_src1 = (const float*)d_in[3];
  const float*      att_dst1 = (const float*)d_in[4];
  const float*      bias1    = (const float*)d_in[5];
  const float*      W2       = (const float*)d_in[6];
  const float*      att_src2 = (const float*)d_in[7];
  const float*      att_dst2 = (const float*)d_in[8];
  const float*      bias2    = (const float*)d_in[9];
  float*            y        = (float*)d_out;

  char* ws = (char*)d_ws;
  size_t off = 0;
  auto carve = [&](size_t bytes) -> void* {
    void* p = ws + off;
    off += (bytes + 255) & ~(size_t)255;
    return p;
  };
  float*    h     = (float*)   carve((size_t)N_NODES * HID * 4);     // h1, then h2
  float*    feat  = (float*)   carve((size_t)N_NODES * HID * 4);     // out1 accum -> elu feat
  float*    out2  = (float*)   carve((size_t)N_NODES * OUT_CH * 4);
  float*    a_src = (float*)   carve((size_t)N_NODES * HEADS * 4);
  float*    a_dst = (float*)   carve((size_t)N_NODES * HEADS * 4);
  unsigned* amax  = (unsigned*)carve((size_t)N_NODES * HEADS * 4);
  float*    denom = (float*)   carve((size_t)N_NODES * HEADS * 4);
  int*      src32 = (int*)     carve((size_t)N_EDGES * 4);
  int*      dst32 = (int*)     carve((size_t)N_EDGES * 4);

  const int EBLK  = (N_EDGES + 255) / 256;
  const int EWAVE = (N_EDGES + 7) / 8;      // 1 wave per edge, 8 waves/block
  const int NHBLK = (N_NODES * HEADS + 255) / 256;
  const int GBLK  = (N_NODES + 127) / 128;

  conv_edges<<<EBLK, 256, 0, stream>>>(ei, src32, dst32);

  // ---------------- layer 1 (concat) ----------------
  gat_gemm<<<GBLK, 256, 0, stream>>>(x, W1, h, N_NODES, IN_CH);
  gat_prep<<<NHBLK, 256, 0, stream>>>(h, att_src1, att_dst1, a_src, a_dst, amax, denom);
  zero_f<<<(N_NODES * HID + 255) / 256, 256, 0, stream>>>(feat, N_NODES * HID);
  gat_edge_max<<<EBLK, 256, 0, stream>>>(src32, dst32, a_src, a_dst, amax);
  gat_edge_sum<<<EBLK, 256, 0, stream>>>(src32, dst32, a_src, a_dst, amax, denom);
  gat_edge_agg_concat<<<EWAVE, 256, 0, stream>>>(src32, dst32, a_src, a_dst, amax, denom, h, feat);
  gat_bias_elu<<<(N_NODES * HID + 255) / 256, 256, 0, stream>>>(feat, bias1, N_NODES * HID);

  // ---------------- layer 2 (mean) ----------------
  gat_gemm<<<GBLK, 256, 0, stream>>>(feat, W2, h, N_NODES, HID);
  gat_prep<<<NHBLK, 256, 0, stream>>>(h, att_src2, att_dst2, a_src, a_dst, amax, denom);
  zero_f<<<(N_NODES * OUT_CH + 255) / 256, 256, 0, stream>>>(out2, N_NODES * OUT_CH);
  gat_edge_max<<<EBLK, 256, 0, stream>>>(src32, dst32, a_src, a_dst, amax);
  gat_edge_sum<<<EBLK, 256, 0, stream>>>(src32, dst32, a_src, a_dst, amax, denom);
  gat_edge_agg_mean<<<EWAVE, 256, 0, stream>>>(src32, dst32, a_src, a_dst, amax, denom, h, out2);
  gat_logsoftmax<<<(N_NODES + 255) / 256, 256, 0, stream>>>(out2, bias2, y);
}